// GATLayer_59777354826466
// MI455X (gfx1250) — compile-verified
//
#include <hip/hip_runtime.h>

typedef float v2f __attribute__((ext_vector_type(2)));
typedef float v8f __attribute__((ext_vector_type(8)));

#define C_IN   256
#define C_HEAD 128
#define C_OUT  256
#define ALPHA  0.2f

// ---------------------------------------------------------------------------
// GEMM: H[N x 256] = X[N x 256] @ [W1 | W2]  (each W is 256 x 128, row-major)
// One wave computes a 64x16 strip of H using V_WMMA_F32_16X16X4_F32.
// A (16x4 f32): lane<16 holds {K=k,K=k+1} of row (lane&15); lane>=16 holds {K=k+2,K=k+3}.
// B (4x16 f32): VGPR0 = rows {k | k+2}, VGPR1 = rows {k+1 | k+3}, N = lane&15.
// C/D (16x16 f32, 8 VGPRs): VGPR v -> row v (lanes 0-15) / row v+8 (lanes 16-31).
// ---------------------------------------------------------------------------
__global__ __launch_bounds__(128) void gat_gemm(
    const float* __restrict__ X, const float* __restrict__ W1,
    const float* __restrict__ W2, float* __restrict__ H, int N)
{
    const int lane   = threadIdx.x & 31;
    const int waveId = threadIdx.x >> 5;
    const int task   = blockIdx.x * 4 + waveId;      // (m64, n16) tile per wave
    const int nTiles = C_OUT / 16;                   // 16
    const int m64    = task / nTiles;
    const int n16    = task % nTiles;
    const int row0   = m64 * 64;
    if (row0 >= N) return;

    const int col   = n16 * 16 + (lane & 15);
    const float* Wh = (col < C_HEAD) ? W1 : W2;
    const int  wcol = (col < C_HEAD) ? col : (col - C_HEAD);
    const int  koff = (lane >= 16) ? 2 : 0;
    const int  mloc = lane & 15;

    // clamp tail rows for loads (stores are guarded / fast-pathed)
    int r0 = row0 +      mloc; if (r0 >= N) r0 = N - 1;
    int r1 = row0 + 16 + mloc; if (r1 >= N) r1 = N - 1;
    int r2 = row0 + 32 + mloc; if (r2 >= N) r2 = N - 1;
    int r3 = row0 + 48 + mloc; if (r3 >= N) r3 = N - 1;
    const float* x0 = X + (size_t)r0 * C_IN + koff;
    const float* x1 = X + (size_t)r1 * C_IN + koff;
    const float* x2 = X + (size_t)r2 * C_IN + koff;
    const float* x3 = X + (size_t)r3 * C_IN + koff;
    const float* wp = Wh + (size_t)koff * C_HEAD + wcol;

    v8f acc0 = {}, acc1 = {}, acc2 = {}, acc3 = {};

#pragma unroll 4
    for (int k = 0; k < C_IN; k += 4) {
        v2f b;
        b.x = wp[(size_t)k * C_HEAD];                 // row k+koff
        b.y = wp[(size_t)(k + 1) * C_HEAD];           // row k+koff+1
        v2f a0 = *(const v2f*)(x0 + k);
        v2f a1 = *(const v2f*)(x1 + k);
        v2f a2 = *(const v2f*)(x2 + k);
        v2f a3 = *(const v2f*)(x3 + k);
        acc0 = __builtin_amdgcn_wmma_f32_16x16x4_f32(false, a0, false, b, (short)0, acc0, false, false);
        acc1 = __builtin_amdgcn_wmma_f32_16x16x4_f32(false, a1, false, b, (short)0, acc1, false, false);
        acc2 = __builtin_amdgcn_wmma_f32_16x16x4_f32(false, a2, false, b, (short)0, acc2, false, false);
        acc3 = __builtin_amdgcn_wmma_f32_16x16x4_f32(false, a3, false, b, (short)0, acc3, false, false);
    }

    const int mAdd = (lane < 16) ? 0 : 8;
    const int cIdx = n16 * 16 + (lane & 15);

    if (row0 + 64 <= N) {
        // fast path (781 of 782 strips): unguarded coalesced stores
        float* h0 = H + (size_t)(row0 +      mAdd) * C_OUT + cIdx;
        float* h1 = H + (size_t)(row0 + 16 + mAdd) * C_OUT + cIdx;
        float* h2 = H + (size_t)(row0 + 32 + mAdd) * C_OUT + cIdx;
        float* h3 = H + (size_t)(row0 + 48 + mAdd) * C_OUT + cIdx;
#pragma unroll
        for (int v = 0; v < 8; ++v) {
            h0[(size_t)v * C_OUT] = acc0[v];
            h1[(size_t)v * C_OUT] = acc1[v];
            h2[(size_t)v * C_OUT] = acc2[v];
            h3[(size_t)v * C_OUT] = acc3[v];
        }
    } else {
#pragma unroll
        for (int v = 0; v < 8; ++v) {
            int rr0 = row0 +      v + mAdd;
            int rr1 = row0 + 16 + v + mAdd;
            int rr2 = row0 + 32 + v + mAdd;
            int rr3 = row0 + 48 + v + mAdd;
            if (rr0 < N) H[(size_t)rr0 * C_OUT + cIdx] = acc0[v];
            if (rr1 < N) H[(size_t)rr1 * C_OUT + cIdx] = acc1[v];
            if (rr2 < N) H[(size_t)rr2 * C_OUT + cIdx] = acc2[v];
            if (rr3 < N) H[(size_t)rr3 * C_OUT + cIdx] = acc3[v];
        }
    }
}

// ---------------------------------------------------------------------------
// Per-node attention scalars: S[0..N)=h1.aL1, [N..2N)=h1.aR1, [2N..3N)=h2.aL2,
// [3N..4N)=h2.aR2.  One wave (32 lanes x float4) per node; butterfly reduce.
// ---------------------------------------------------------------------------
__global__ __launch_bounds__(256) void gat_scores(
    const float* __restrict__ H, const float* __restrict__ a1,
    const float* __restrict__ a2, float* __restrict__ S, int N)
{
    const int lane = threadIdx.x & 31;
    const int n    = blockIdx.x * 8 + (threadIdx.x >> 5);
    if (n >= N) return;
    const int c = lane * 4;

    float4 h1  = *(const float4*)(H + (size_t)n * C_OUT + c);
    float4 h2  = *(const float4*)(H + (size_t)n * C_OUT + C_HEAD + c);
    float4 al1 = *(const float4*)(a1 + c);
    float4 ar1 = *(const float4*)(a1 + C_HEAD + c);
    float4 al2 = *(const float4*)(a2 + c);
    float4 ar2 = *(const float4*)(a2 + C_HEAD + c);

    float p0 = h1.x*al1.x + h1.y*al1.y + h1.z*al1.z + h1.w*al1.w;
    float p1 = h1.x*ar1.x + h1.y*ar1.y + h1.z*ar1.z + h1.w*ar1.w;
    float p2 = h2.x*al2.x + h2.y*al2.y + h2.z*al2.z + h2.w*al2.w;
    float p3 = h2.x*ar2.x + h2.y*ar2.y + h2.z*ar2.z + h2.w*ar2.w;

#pragma unroll
    for (int off = 16; off > 0; off >>= 1) {
        p0 += __shfl_xor(p0, off, 32);
        p1 += __shfl_xor(p1, off, 32);
        p2 += __shfl_xor(p2, off, 32);
        p3 += __shfl_xor(p3, off, 32);
    }
    if (lane == 0) {
        S[n]                 = p0;
        S[(size_t)N + n]     = p1;
        S[(size_t)2 * N + n] = p2;
        S[(size_t)3 * N + n] = p3;
    }
}

// ---------------------------------------------------------------------------
// Edge scatter: one wave per edge.  w = exp(-leaky_relu(sL[src]+sR[dst])).
// Gather H[dst] row (L2-resident) and scatter w*h into acc[src] with hardware
// fp32 atomics (global_atomic_add_f32); lane 0 accumulates row sums.
// ---------------------------------------------------------------------------
__global__ __launch_bounds__(256) void gat_edges(
    const float* __restrict__ H, const float* __restrict__ S,
    const int* __restrict__ ei, float* __restrict__ acc,
    float* __restrict__ rs, int N, int E)
{
    const int lane   = threadIdx.x & 31;
    const int wave   = (blockIdx.x * blockDim.x + threadIdx.x) >> 5;
    const int nWaves = (gridDim.x * blockDim.x) >> 5;
    const int c      = lane * 4;

    for (int e = wave; e < E; e += nWaves) {
        const int src = ei[e];
        const int dst = ei[(size_t)E + e];

        const float l1  = S[src] + S[(size_t)N + dst];
        const float l2  = S[(size_t)2 * N + src] + S[(size_t)3 * N + dst];
        const float lr1 = (l1 > 0.f) ? l1 : ALPHA * l1;
        const float lr2 = (l2 > 0.f) ? l2 : ALPHA * l2;
        const float w1  = __expf(-lr1);
        const float w2  = __expf(-lr2);

        const float4 h1 = *(const float4*)(H + (size_t)dst * C_OUT + c);
        const float4 h2 = *(const float4*)(H + (size_t)dst * C_OUT + C_HEAD + c);

        float* p1 = acc + (size_t)src * C_OUT + c;
        unsafeAtomicAdd(p1 + 0, w1 * h1.x);
        unsafeAtomicAdd(p1 + 1, w1 * h1.y);
        unsafeAtomicAdd(p1 + 2, w1 * h1.z);
        unsafeAtomicAdd(p1 + 3, w1 * h1.w);
        float* p2 = p1 + C_HEAD;
        unsafeAtomicAdd(p2 + 0, w2 * h2.x);
        unsafeAtomicAdd(p2 + 1, w2 * h2.y);
        unsafeAtomicAdd(p2 + 2, w2 * h2.z);
        unsafeAtomicAdd(p2 + 3, w2 * h2.w);
        if (lane == 0) {
            unsafeAtomicAdd(rs + src, w1);
            unsafeAtomicAdd(rs + (size_t)N + src, w2);
        }
    }
}

// ---------------------------------------------------------------------------
// Finalize in place: out = elu(acc / rowsum)   (head 0 cols 0-127, head 1 128-255)
// ---------------------------------------------------------------------------
__global__ __launch_bounds__(256) void gat_final(
    float* __restrict__ out, const float* __restrict__ rs, int N)
{
    const long long total = (long long)N * C_OUT;
    for (long long i = blockIdx.x * (long long)blockDim.x + threadIdx.x;
         i < total; i += (long long)gridDim.x * blockDim.x) {
        const int n    = (int)(i >> 8);
        const int ccol = (int)(i & 255);
        const float denom = (ccol < C_HEAD) ? rs[n] : rs[(size_t)N + n];
        const float v = out[i] / denom;
        out[i] = (v > 0.f) ? v : (__expf(v) - 1.f);
    }
}

extern "C" void kernel_launch(void* const* d_in, const int* in_sizes, int n_in,
                              void* d_out, int out_size, void* d_ws, size_t ws_size,
                              hipStream_t stream)
{
    const float* X  = (const float*)d_in[0];
    const float* W1 = (const float*)d_in[1];
    const float* a1 = (const float*)d_in[2];
    const float* W2 = (const float*)d_in[3];
    const float* a2 = (const float*)d_in[4];
    const int*   ei = (const int*)d_in[5];

    const int cHead = in_sizes[2] / 2;            // 128
    const int cIn   = in_sizes[1] / cHead;        // 256
    const int N     = in_sizes[0] / cIn;          // 50000
    const int E     = in_sizes[5] / 2;            // 800000
    (void)cHead; (void)cIn; (void)n_in; (void)out_size; (void)ws_size;

    float* out = (float*)d_out;
    float* H   = (float*)d_ws;                    // N*256 f32
    float* S   = H + (size_t)N * C_OUT;           // 4*N  f32
    float* rs  = S + (size_t)4 * N;               // 2*N  f32

    hipMemsetAsync(out, 0, (size_t)N * C_OUT * sizeof(float), stream);
    hipMemsetAsync(rs,  0, (size_t)2 * N * sizeof(float), stream);

    const int mTiles   = (N + 63) / 64;
    const int tasks    = mTiles * (C_OUT / 16);
    const int gemmBlks = (tasks + 3) / 4;         // 4 waves per 128-thread block
    gat_gemm<<<gemmBlks, 128, 0, stream>>>(X, W1, W2, H, N);

    gat_scores<<<(N + 7) / 8, 256, 0, stream>>>(H, a1, a2, S, N);

    int edgeBlks = (E + 7) / 8;
    if (edgeBlks > 4096) edgeBlks = 4096;         // grid-stride over edges
    gat_edges<<<edgeBlks, 256, 0, stream>>>(H, S, ei, out, rs, N, E);

    const long long total = (long long)N * C_OUT;
    int finBlks = (int)((total + 255) / 256);
    if (finBlks > 16384) finBlks = 16384;
    gat_final<<<finBlks, 256, 0, stream>>>(out, rs, N);
}